// TipletMiningLoss_76192719831357
// MI455X (gfx1250) — compile-verified
//
#include <hip/hip_runtime.h>
#include <hip/hip_bf16.h>

typedef float v2f __attribute__((ext_vector_type(2)));
typedef float v8f __attribute__((ext_vector_type(8)));

#define WAVES 4
#define BSTRIDE 132                 // padded row stride (floats): conflict-free ds_load_b64 B reads
#define SLAB_FLOATS (16 * BSTRIDE)  // one 16-row column slab
#define SLAB_BYTES (SLAB_FLOATS * 4)
#define EPSV 1e-8f
#define USE_ASYNC_LDS 1

// ---------------- kernel 0: per-row squared norm and sum (warp per row) -------------
__global__ void row_stats_kernel(const float* __restrict__ x, float* __restrict__ sq,
                                 float* __restrict__ s, float* __restrict__ acc, int N) {
  const int warp = (int)((blockIdx.x * blockDim.x + threadIdx.x) >> 5);
  const int lane = threadIdx.x & 31;
  if (blockIdx.x == 0 && threadIdx.x == 0) acc[0] = 0.0f;
  if (warp >= N) return;
  const float4 v = *(const float4*)(x + (size_t)warp * 128 + lane * 4);
  float ssq = v.x * v.x + v.y * v.y + v.z * v.z + v.w * v.w;
  float ss  = v.x + v.y + v.z + v.w;
#pragma unroll
  for (int m = 16; m >= 1; m >>= 1) {
    ssq += __shfl_xor(ssq, m, 32);
    ss  += __shfl_xor(ss,  m, 32);
  }
  if (lane == 0) { sq[warp] = ssq; s[warp] = ss; }
}

// Async-DMA one 16x128 f32 row slab (rows colBase..colBase+15) into LDS at ldsBase.
// Per instruction: 32 lanes x 16B = one 512B row. ASYNCcnt += 16.
__device__ __forceinline__ void prefetch_slab_async(const float* __restrict__ x,
                                                    int colBase, unsigned ldsBase,
                                                    int lane) {
  const unsigned lbyte = ldsBase + (unsigned)lane * 16u;
  const unsigned gbyte = (unsigned)colBase * 512u + (unsigned)lane * 16u;
#pragma unroll
  for (int t = 0; t < 16; ++t) {
    const unsigned l = lbyte + (unsigned)t * (unsigned)(BSTRIDE * 4);
    const unsigned g = gbyte + (unsigned)t * 512u;
#if USE_ASYNC_LDS
    asm volatile("global_load_async_to_lds_b128 %0, %1, %2"
                 :: "v"(l), "v"(g), "s"(x) : "memory");
#else
    (void)l; (void)g;
#endif
  }
}

// ---------------- kernel 1: fused WMMA gram tile + masked row max/min ---------------
__global__ __launch_bounds__(128, 1)
void triplet_tile_kernel(const float* __restrict__ x, const int* __restrict__ lab,
                         const float* __restrict__ sq, const float* __restrict__ s,
                         float* __restrict__ acc, int N) {
  extern __shared__ float smem[];
  float* bufs = smem;                                   // WAVES * 2 * SLAB_FLOATS
  float* posW = smem + WAVES * 2 * SLAB_FLOATS;         // WAVES * 16
  float* negW = posW + WAVES * 16;                      // WAVES * 16

  const int tid    = threadIdx.x;
  const int wave   = tid >> 5;
  const int lane   = tid & 31;
  const int lane16 = lane & 15;
  const int hi     = lane >> 4;
  const int koff   = hi * 2;           // fp32 16x16x4 layout: lane halves hold K {0,1}/{2,3}
  const int mBase  = blockIdx.x * 16;
  const int nTiles = N >> 4;

  float* buf0 = bufs + wave * 2 * SLAB_FLOATS;
  float* buf1 = buf0 + SLAB_FLOATS;
  const unsigned ldsB0 = (unsigned)(size_t)buf0;        // low 32 bits == LDS byte offset
  const unsigned ldsB1 = (unsigned)(size_t)buf1;

  // Hoisted A fragments: row (mBase+lane16), float2 at k*4+koff (ISA 16x4 f32 A layout)
  v2f aF[32];
  const float* arow = x + (size_t)(mBase + lane16) * 128 + koff;
#pragma unroll
  for (int kk = 0; kk < 32; ++kk) {
    float2 t = *(const float2*)(arow + kk * 4);
    aF[kk].x = t.x; aF[kk].y = t.y;
  }

  float rowSq[8], rowS[8]; int rowLab[8];
#pragma unroll
  for (int v = 0; v < 8; ++v) {
    const int i = mBase + v + 8 * hi;
    rowSq[v] = sq[i]; rowS[v] = s[i]; rowLab[v] = lab[i];
  }

  float posAcc[8], negAcc[8];
#pragma unroll
  for (int v = 0; v < 8; ++v) { posAcc[v] = 0.0f; negAcc[v] = 3.0e38f; }

#if !USE_ASYNC_LDS
  // synchronous fallback staging
  auto stage_sync = [&](int colBase, float* dstB) {
#pragma unroll
    for (int t = 0; t < 16; ++t) {
      const float4 v4 = *(const float4*)(x + (size_t)(colBase + t) * 128 + lane * 4);
      *(float4*)(&dstB[t * BSTRIDE + lane * 4]) = v4;
    }
  };
#endif

  // Prologue: prefetch first tile into buffer 0
#if USE_ASYNC_LDS
  prefetch_slab_async(x, wave * 16, ldsB0, lane);
#endif

  int parity = 0;
  for (int ct = wave; ct < nTiles; ct += WAVES) {
    float* myB = parity ? buf1 : buf0;
#if USE_ASYNC_LDS
    // Prefetch next tile (dummy re-prefetch of current on last iter keeps wait uniform)
    int ctNext = ct + WAVES; if (ctNext >= nTiles) ctNext = ct;
    prefetch_slab_async(x, ctNext * 16, parity ? ldsB0 : ldsB1, lane);
    // <=16 outstanding ==> current slab's 16 async ops have completed (in-order)
    asm volatile("s_wait_asynccnt 0x10" ::: "memory");
#else
    stage_sync(ct * 16, myB);
#endif

    // 16x16 gram tile: 4 independent V_WMMA_F32_16X16X4_F32 accumulation chains
    v8f c0 = {}, c1 = {}, c2 = {}, c3 = {};
#pragma unroll
    for (int kk = 0; kk < 32; kk += 4) {
      const float* bp = &myB[lane16 * BSTRIDE + koff];
      float2 t0 = *(const float2*)(bp + (kk + 0) * 4);
      float2 t1 = *(const float2*)(bp + (kk + 1) * 4);
      float2 t2 = *(const float2*)(bp + (kk + 2) * 4);
      float2 t3 = *(const float2*)(bp + (kk + 3) * 4);
      v2f b0; b0.x = t0.x; b0.y = t0.y;
      v2f b1; b1.x = t1.x; b1.y = t1.y;
      v2f b2; b2.x = t2.x; b2.y = t2.y;
      v2f b3; b3.x = t3.x; b3.y = t3.y;
      c0 = __builtin_amdgcn_wmma_f32_16x16x4_f32(false, aF[kk + 0], false, b0, (short)0, c0, false, false);
      c1 = __builtin_amdgcn_wmma_f32_16x16x4_f32(false, aF[kk + 1], false, b1, (short)0, c1, false, false);
      c2 = __builtin_amdgcn_wmma_f32_16x16x4_f32(false, aF[kk + 2], false, b2, (short)0, c2, false, false);
      c3 = __builtin_amdgcn_wmma_f32_16x16x4_f32(false, aF[kk + 3], false, b3, (short)0, c3, false, false);
    }
    const v8f c = (c0 + c1) + (c2 + c3);

    // Epilogue: distance + masks + running row max/min (dist matrix never materialized)
    const int   colBase = ct * 16;
    const int   j   = colBase + lane16;
    const float sqj = sq[j], sj = s[j];
    const int  labj = lab[j];
#pragma unroll
    for (int v = 0; v < 8; ++v) {
      const int i = mBase + v + 8 * hi;
      float d2 = rowSq[v] + sqj - 2.0f * c[v]
               + 2.0f * EPSV * (rowS[v] - sj) + 128.0f * EPSV * EPSV;
      const float dist = sqrtf(fmaxf(d2, 0.0f));
      const bool same = (rowLab[v] == labj);
      if (same && (i != j)) posAcc[v] = fmaxf(posAcc[v], dist);   // furthest positive
      if (!same)            negAcc[v] = fminf(negAcc[v], dist);   // closest negative
    }
    parity ^= 1;
  }

  // Reduce across the 16 lanes of each half (columns of the tile)
#pragma unroll
  for (int v = 0; v < 8; ++v) {
    float p = posAcc[v], n = negAcc[v];
#pragma unroll
    for (int m = 8; m >= 1; m >>= 1) {
      p = fmaxf(p, __shfl_xor(p, m, 32));
      n = fminf(n, __shfl_xor(n, m, 32));
    }
    if (lane16 == 0) { posW[wave * 16 + 8 * hi + v] = p; negW[wave * 16 + 8 * hi + v] = n; }
  }
  __syncthreads();

  // Combine across waves, hinge loss for this block's 16 rows, atomic sum
  if (tid < 16) {
    float p = posW[tid], n = negW[tid];
#pragma unroll
    for (int w = 1; w < WAVES; ++w) {
      p = fmaxf(p, posW[w * 16 + tid]);
      n = fminf(n, negW[w * 16 + tid]);
    }
    float tl = fmaxf(p - n + 1.0f, 0.0f);
#pragma unroll
    for (int m = 8; m >= 1; m >>= 1) tl += __shfl_xor(tl, m, 32);
    if (tid == 0) atomicAdd(acc, tl);
  }
}

// ---------------- kernel 2: mean ----------------------------------------------------
__global__ void finalize_kernel(const float* __restrict__ acc, float* __restrict__ out, int N) {
  if (threadIdx.x == 0 && blockIdx.x == 0) out[0] = acc[0] / (float)N;
}

extern "C" void kernel_launch(void* const* d_in, const int* in_sizes, int n_in,
                              void* d_out, int out_size, void* d_ws, size_t ws_size,
                              hipStream_t stream) {
  const float* x  = (const float*)d_in[0];
  const int* lab  = (const int*)d_in[1];
  const int N     = in_sizes[1];          // 8192; D fixed at 128 by reference
  float* wsf = (float*)d_ws;
  float* acc = wsf;                        // [0] loss accumulator
  float* sq  = wsf + 8;                    // row squared norms
  float* s   = wsf + 8 + N;                // row sums
  float* out = (float*)d_out;

  const size_t lds_bytes = (size_t)(WAVES * 2 * SLAB_FLOATS + 2 * WAVES * 16) * sizeof(float);

  row_stats_kernel<<<N / 8, 256, 0, stream>>>(x, sq, s, acc, N);
  triplet_tile_kernel<<<N / 16, 128, lds_bytes, stream>>>(x, lab, sq, s, acc, N);
  finalize_kernel<<<1, 1, 0, stream>>>(acc, out, N);
}